// Encoder_62998580298235
// MI455X (gfx1250) — compile-verified
//
#include <hip/hip_runtime.h>
#include <hip/hip_bf16.h>
#include <stdint.h>

typedef __attribute__((ext_vector_type(2))) float v2f;
typedef __attribute__((ext_vector_type(8))) float v8f;

#define IN_DIM 128
#define HID_DIM 128
#define KDIM 256   // 2 * IN_DIM

// ---------------- zero workspace ----------------
__global__ void zero_kernel(uint32_t* __restrict__ p, long n) {
    long i = (long)blockIdx.x * blockDim.x + threadIdx.x;
    if (i < n) p[i] = 0u;
}

// ---------------- in-degree (integer HW atomics) ----------------
__global__ void degree_kernel(const long long* __restrict__ dst,
                              uint32_t* __restrict__ deg, int E) {
    int e = blockIdx.x * blockDim.x + threadIdx.x;
    if (e < E) atomicAdd(&deg[(int)dst[e]], 1u);
}

// ---------------- norm = clip(deg,1)^-0.5 ----------------
__global__ void norm_kernel(const uint32_t* __restrict__ deg,
                            float* __restrict__ nrm, int N) {
    int i = blockIdx.x * blockDim.x + threadIdx.x;
    if (i < N) {
        float d = (float)deg[i];
        nrm[i] = 1.0f / sqrtf(fmaxf(d, 1.0f));
    }
}

// ---------------- edge scatter: agg[dst] += h[src]*norm[src] ----------------
// One wave32 per edge: 32 lanes x float4 = 128 floats. h/agg are L2-resident
// (51 MB each in 192 MB L2); uses hardware global_atomic_add_f32.
__global__ void scatter_kernel(const float* __restrict__ h,
                               const long long* __restrict__ src,
                               const long long* __restrict__ dst,
                               const float* __restrict__ nrm,
                               float* __restrict__ agg, int E) {
    int lane = threadIdx.x & 31;
    int e = blockIdx.x * (blockDim.x >> 5) + (threadIdx.x >> 5);
    if (e >= E) return;
    int s = (int)src[e];
    int d = (int)dst[e];
    float ns = nrm[s];
    const float4* h4 = (const float4*)(h + (long)s * IN_DIM);
    float4 v = h4[lane];
    float* base = agg + (long)d * IN_DIM + lane * 4;
    unsafeAtomicAdd(base + 0, v.x * ns);
    unsafeAtomicAdd(base + 1, v.y * ns);
    unsafeAtomicAdd(base + 2, v.z * ns);
    unsafeAtomicAdd(base + 3, v.w * ns);
}

// ---------------- fused GEMM + bias + L2-normalize ----------------
// Block = 256 threads = 8 wave32. Block owns 16 output rows; wave w owns
// output cols [16w,16w+16). f32 WMMA 16x16x4, K = 256 -> 64 wmma per wave.
// A-frag (ISA 16x4 f32 layout): lane m=lane&15, K = k0 + (lane<16?0:2) + j.
// B-frag symmetric: col n0+(lane&15), same K indexing.
__launch_bounds__(256)
__global__ void tag_gemm_kernel(const float* __restrict__ h,
                                const float* __restrict__ agg,
                                const float* __restrict__ nrm,
                                const float* __restrict__ W,
                                const float* __restrict__ bias,
                                float* __restrict__ out, int N) {
    __shared__ float fs[16][258];   // fstack tile [16 rows][256 K], padded
    __shared__ float ot[16][132];   // pre-norm output tile, padded
    __shared__ float rowsum[16];

    int tid  = threadIdx.x;
    int row0 = blockIdx.x * 16;

    // stage fstack = [h | agg*norm] for 16 rows into LDS
    for (int t = tid; t < 16 * KDIM; t += 256) {
        int r = t >> 8;
        int c = t & 255;
        int node = row0 + r;
        float v = 0.0f;
        if (node < N) {
            if (c < IN_DIM) v = h[(long)node * IN_DIM + c];
            else            v = agg[(long)node * IN_DIM + (c - IN_DIM)] * nrm[node];
        }
        fs[r][c] = v;
    }
    if (tid < 16) rowsum[tid] = 0.0f;
    __syncthreads();

    int wave = tid >> 5;
    int lane = tid & 31;
    int m    = lane & 15;
    int koff = (lane < 16) ? 0 : 2;
    int ncol = wave * 16 + m;

    v8f acc = {};
    for (int k0 = 0; k0 < KDIM; k0 += 4) {
        v2f a, b2;
        a[0]  = fs[m][k0 + koff];
        a[1]  = fs[m][k0 + koff + 1];
        b2[0] = W[(k0 + koff) * HID_DIM + ncol];
        b2[1] = W[(k0 + koff + 1) * HID_DIM + ncol];
        acc = __builtin_amdgcn_wmma_f32_16x16x4_f32(
            /*neg_a=*/false, a, /*neg_b=*/false, b2,
            /*c_mod=*/(short)0, acc, /*reuse_a=*/false, /*reuse_b=*/false);
    }

    // epilogue: bias, stash tile (C/D layout: vgpr r -> row r (lanes 0-15)
    // or row r+8 (lanes 16-31), col = ncol)
    float bv = bias[ncol];
    int rbase = (lane < 16) ? 0 : 8;
    #pragma unroll
    for (int r = 0; r < 8; ++r) ot[rbase + r][ncol] = acc[r] + bv;
    __syncthreads();

    // row sum of squares (16 threads per row, ds_add_f32 accumulate)
    {
        int r = tid >> 4;
        int seg = tid & 15;
        float p = 0.0f;
        #pragma unroll
        for (int i = 0; i < 8; ++i) {
            float v = ot[r][seg * 8 + i];
            p += v * v;
        }
        atomicAdd(&rowsum[r], p);
    }
    __syncthreads();

    // normalize and store
    for (int t = tid; t < 16 * HID_DIM; t += 256) {
        int r = t >> 7;
        int c = t & 127;
        int node = row0 + r;
        if (node < N)
            out[(long)node * HID_DIM + c] = ot[r][c] / sqrtf(rowsum[r]);
    }
}

extern "C" void kernel_launch(void* const* d_in, const int* in_sizes, int n_in,
                              void* d_out, int out_size, void* d_ws, size_t ws_size,
                              hipStream_t stream) {
    const float*     h    = (const float*)d_in[0];
    const long long* src  = (const long long*)d_in[1];
    const long long* dst  = (const long long*)d_in[2];
    const float*     W    = (const float*)d_in[3];
    const float*     bias = (const float*)d_in[4];
    float*           out  = (float*)d_out;

    int N = in_sizes[0] / IN_DIM;   // 100000
    int E = in_sizes[1];            // 3200000

    // ws layout: agg [N*128 f32] | nrm [N f32] | deg [N u32]
    float*    agg = (float*)d_ws;
    float*    nrm = agg + (size_t)N * IN_DIM;
    uint32_t* deg = (uint32_t*)(nrm + N);

    long nwords = (long)N * (IN_DIM + 2);   // zero agg + nrm + deg
    zero_kernel<<<(int)((nwords + 255) / 256), 256, 0, stream>>>((uint32_t*)d_ws, nwords);

    degree_kernel<<<(E + 255) / 256, 256, 0, stream>>>(dst, deg, E);
    norm_kernel<<<(N + 255) / 256, 256, 0, stream>>>(deg, nrm, N);

    // one wave per edge, 8 waves per 256-thread block
    scatter_kernel<<<(E + 7) / 8, 256, 0, stream>>>(h, src, dst, nrm, agg, E);

    tag_gemm_kernel<<<(N + 15) / 16, 256, 0, stream>>>(h, agg, nrm, W, bias, out, N);
}